// StructuredBackBone8xMRS2_22428319220763
// MI455X (gfx1250) — compile-verified
//
#include <hip/hip_runtime.h>

// MI455X / gfx1250: wave32, WMMA 16x16x4 f32 + Tensor Data Mover staging.
typedef __attribute__((ext_vector_type(2))) float v2f;
typedef __attribute__((ext_vector_type(8))) float v8f;
typedef __attribute__((ext_vector_type(4))) unsigned u32x4;
typedef __attribute__((ext_vector_type(8))) unsigned u32x8;

static constexpr int kBatch = 4;
static constexpr int kNl    = 2000;   // low-res points per batch
static constexpr int kC     = 16;     // hr feature channels
static constexpr int kIn    = 19;     // 3 rel-xyz + 16 feat
static constexpr int kFeatC = 128;    // fused channel count
static constexpr int kOutC  = 64;     // final output channels
static constexpr int kStagePts = 2048;  // hr points per TDM stage (24 KB)

// ---------------------------------------------------------------------------
// Tensor Data Mover: async 1-D f32 tile, global -> LDS. One issue per wave.
// D# group0: count=1 | lds_addr | global_addr | type=2.
// D# group1: data_size=4B; tensor_dim0 = tile_dim0 = nElems (1-D tile);
//            tensor_dim1=1, tile_dim1/2 = 0 (unused); no mask/pad/iterate.
// Tracked on TENSORcnt (wait via s_wait_tensorcnt). nElems must be <= 65535.
// ---------------------------------------------------------------------------
__device__ __forceinline__ void tdm_load_f32_1d(unsigned lds_addr,
                                                const float* __restrict__ src,
                                                unsigned nElems)
{
  unsigned long long ga = (unsigned long long)(size_t)src;
  u32x4 g0;
  g0.x = 1u;                                   // count=1 (user mode)
  g0.y = lds_addr;                             // LDS byte address
  g0.z = (unsigned)ga;                         // global_addr[31:0]
  g0.w = (unsigned)(ga >> 32) | (2u << 30);    // global_addr[56:32] | type=2
  u32x8 g1;
  g1[0] = 2u << 16;                            // data_size = 4 bytes
  g1[1] = (nElems & 0xFFFFu) << 16;            // tensor_dim0[15:0]
  g1[2] = ((nElems >> 16) & 0xFFFFu) | (1u << 16); // tensor_dim0[31:16] | tensor_dim1=1
  g1[3] = (nElems & 0xFFFFu) << 16;            // tile_dim0 = nElems
  g1[4] = 0u;                                  // tile_dim1/2 unused
  g1[5] = nElems;                              // tensor_dim0_stride (don't-care, 1-D)
  g1[6] = 0u;
  g1[7] = 0u;
  asm volatile("tensor_load_to_lds %0, %1" :: "s"(g0), "s"(g1) : "memory");
}

// ---------------------------------------------------------------------------
// Per-candidate MLP contribution, max-pooled into LDS accumulator.
// One wave processes one candidate; lanes spread over output channels.
// ReLU output >= 0, so float-bit atomicMax against 0-initialized bits is exact.
// ---------------------------------------------------------------------------
template<int OUT>
__device__ __forceinline__ void contrib_maxpool(
    int m, int j, int gb, int Nh,
    const float* __restrict__ hr_xyz, const float* __restrict__ hr_feat,
    const float* __restrict__ W, const float* __restrict__ bias,
    const float* lx, const float* ly, const float* lz,
    unsigned* hbits, int lane)
{
  const float* hp = hr_xyz + (size_t)(gb * Nh + j) * 3;
  const float* fp = hr_feat + (size_t)(gb * Nh + j) * kC;
  float gx = hp[0] - lx[m];
  float gy = hp[1] - ly[m];
  float gz = hp[2] - lz[m];
  for (int o = lane; o < OUT; o += 32) {
    const float* w = W + o * kIn;
    float acc = bias[o] + gx * w[0] + gy * w[1] + gz * w[2];
#pragma unroll
    for (int c = 0; c < kC; ++c) acc += fp[c] * w[3 + c];
    float h = fmaxf(acc, 0.0f);
    atomicMax(&hbits[m * OUT + o], __float_as_uint(h));
  }
}

// ---------------------------------------------------------------------------
// Ball-group + 1x1 MLP + maxpool for one scale.
// Grid: (Nl/16, B). Block: 256 = 8 waves. Each block owns 16 lr points.
// hr_xyz streamed by the TDM into a double-buffered LDS tile; distance tiles
// via V_WMMA_F32_16X16X4_F32:
//   A row m  = (-2ax, -2ay, -2az, 1)        (16x4)
//   B col n  = ( bx,   by,   bz, |b|^2)     (4x16)
//   C = A*B  = -2 a.b + |b|^2 ; d2 = C + |a|^2
// ---------------------------------------------------------------------------
template<int OUT, int CAP>
__global__ __launch_bounds__(256)
void grouper_kernel(const float* __restrict__ lr_xyz,
                    const float* __restrict__ hr_xyz,
                    const float* __restrict__ hr_feat,
                    const float* __restrict__ W,
                    const float* __restrict__ bias,
                    float* __restrict__ feats,
                    int Nh, int chOff)
{
  __shared__ float sxyz[2][kStagePts * 3];   // 48 KB TDM double buffer
  __shared__ float lx[16], ly[16], lz[16], la2[16];
  __shared__ unsigned long long minkey[16];
  __shared__ unsigned hbits[16 * OUT];
  __shared__ unsigned cand[CAP];
  __shared__ int ccount;

  const int tid  = threadIdx.x;
  const int lane = tid & 31;
  const int wave = tid >> 5;
  const bool hiL = lane >= 16;
  const int gb   = blockIdx.y;
  const int n0   = blockIdx.x * 16;

  if (tid < 16) {
    const float* p = lr_xyz + (size_t)(gb * kNl + n0 + tid) * 3;
    float x = p[0], y = p[1], z = p[2];
    lx[tid] = x; ly[tid] = y; lz[tid] = z;
    la2[tid] = x * x + y * y + z * z;
    minkey[tid] = ~0ull;
  }
  if (tid == 0) ccount = 0;
  for (int i = tid; i < 16 * OUT; i += 256) hbits[i] = 0u;

  const float* hrb = hr_xyz + (size_t)gb * Nh * 3;
  const int nStages = (Nh + kStagePts - 1) / kStagePts;

  // Kick off stage 0 DMA before we even finish setup.
  if (wave == 0) {
    unsigned e0 = (unsigned)(3 * ((Nh < kStagePts) ? Nh : kStagePts));
    tdm_load_f32_1d((unsigned)(size_t)&sxyz[0][0], hrb, e0);
  }
  __syncthreads();

  // A fragment (constant across the Nh scan).
  // lanes 0-15: (K0,K1) = (-2x,-2y); lanes 16-31: (K2,K3) = (-2z, 1)
  const int am = lane & 15;
  v2f afrag;
  afrag.x = hiL ? (-2.0f * lz[am]) : (-2.0f * lx[am]);
  afrag.y = hiL ? 1.0f             : (-2.0f * ly[am]);

  // rows this lane owns in the C tile: M = r + (hiL ? 8 : 0)
  float a2r[8];
#pragma unroll
  for (int r = 0; r < 8; ++r) a2r[r] = la2[r + (hiL ? 8 : 0)];

  unsigned long long bkey[8];
#pragma unroll
  for (int r = 0; r < 8; ++r) bkey[r] = ~0ull;

  for (int s = 0; s < nStages; ++s) {
    const int base   = s * kStagePts;
    const int scount = (Nh - base < kStagePts) ? (Nh - base) : kStagePts;

    if (wave == 0) {
      if (s + 1 < nStages) {
        const int nb = (s + 1) * kStagePts;
        const int nc2 = (Nh - nb < kStagePts) ? (Nh - nb) : kStagePts;
        tdm_load_f32_1d((unsigned)(size_t)&sxyz[(s + 1) & 1][0],
                        hrb + (size_t)nb * 3, (unsigned)(3 * nc2));
        __builtin_amdgcn_s_wait_tensorcnt(1);  // stage s complete (in-order)
      } else {
        __builtin_amdgcn_s_wait_tensorcnt(0);
      }
    }
    __syncthreads();  // stage s visible to all waves

    const float* sb = sxyz[s & 1];
    // scount is always a multiple of 16 here (Nh % kStagePts in
    // {1664,1856,1952} for the three scales), so no tail divergence.
    for (int jj = wave * 16; jj < scount; jj += 16 * 8) {
      const int jl = jj + (lane & 15);
      const float* bp = sb + jl * 3;
      float hx = bp[0], hy = bp[1], hz = bp[2];
      float h2 = hx * hx + hy * hy + hz * hz;
      v2f bfrag;
      bfrag.x = hiL ? hz : hx;
      bfrag.y = hiL ? h2 : hy;

      v8f c = {};
      c = __builtin_amdgcn_wmma_f32_16x16x4_f32(
          false, afrag, false, bfrag, (short)0, c, false, false);

      const int j = base + jl;
#pragma unroll
      for (int r = 0; r < 8; ++r) {
        float d2 = c[r] + a2r[r];
        if (d2 <= 1.0f) {
          int slot = atomicAdd(&ccount, 1);
          if (slot < CAP)
            cand[slot] = ((unsigned)j << 4) | (unsigned)(r + (hiL ? 8 : 0));
        }
        float dk = fmaxf(d2, 0.0f);  // keep uint-bit key monotonic
        unsigned long long key =
            ((unsigned long long)__float_as_uint(dk) << 32) | (unsigned)j;
        if (key < bkey[r]) bkey[r] = key;
      }
    }

    // Drain our LDS reads before the TDM recycles this buffer.
    asm volatile("s_wait_dscnt 0x0" ::: "memory");
    __syncthreads();
  }

#pragma unroll
  for (int r = 0; r < 8; ++r)
    atomicMin(&minkey[r + (hiL ? 8 : 0)], bkey[r]);
  __syncthreads();

  // In-radius candidates: one wave per candidate, lanes over channels.
  const int nc = min(ccount, CAP);
  for (int ci = wave; ci < nc; ci += 8) {
    unsigned e = cand[ci];
    contrib_maxpool<OUT>((int)(e & 15u), (int)(e >> 4), gb, Nh,
                         hr_xyz, hr_feat, W, bias, lx, ly, lz, hbits, lane);
  }

  // Nearest-point padding (pointnet2 semantics): always contributes.
  for (int m = wave; m < 16; m += 8) {
    int j = (int)(minkey[m] & 0xFFFFFFFFull);
    contrib_maxpool<OUT>(m, j, gb, Nh,
                         hr_xyz, hr_feat, W, bias, lx, ly, lz, hbits, lane);
  }
  __syncthreads();

  for (int i = tid; i < 16 * OUT; i += 256) {
    int m = i / OUT, o = i - m * OUT;
    feats[(size_t)(gb * kNl + n0 + m) * kFeatC + chOff + o] =
        __uint_as_float(hbits[i]);
  }
}

// ---------------------------------------------------------------------------
// Copy lr_feat [8000,32] into channels [0,32) of fused feats [8000,128].
// ---------------------------------------------------------------------------
__global__ __launch_bounds__(256)
void copy_lr_feat_kernel(const float* __restrict__ lr_feat,
                         float* __restrict__ feats)
{
  int i = blockIdx.x * 256 + threadIdx.x;
  if (i < kBatch * kNl * 32) {
    int r = i >> 5, c = i & 31;
    feats[(size_t)r * kFeatC + c] = lr_feat[i];
  }
}

// ---------------------------------------------------------------------------
// Fused 128->64 GEMM + BatchNorm(eval) + ReLU via chained f32 WMMA.
// Grid: 125 blocks x 128 threads (4 waves); wave w handles 16 rows.
// ---------------------------------------------------------------------------
__global__ __launch_bounds__(128)
void out_gemm_kernel(const float* __restrict__ feats,
                     const float* __restrict__ Wout,
                     const float* __restrict__ bn_gamma,
                     const float* __restrict__ bn_beta,
                     const float* __restrict__ bn_mean,
                     const float* __restrict__ bn_var,
                     float* __restrict__ out)
{
  __shared__ float sW[kOutC * kFeatC];        // 32 KB
  __shared__ float sF[4][16 * kFeatC];        // 32 KB

  const int tid  = threadIdx.x;
  const int lane = tid & 31;
  const int wave = tid >> 5;
  const bool hiL = lane >= 16;
  const int row0 = blockIdx.x * 64 + wave * 16;

  for (int i = tid; i < kOutC * kFeatC; i += 128) sW[i] = Wout[i];
  for (int i = lane; i < 16 * kFeatC; i += 32)
    sF[wave][i] = feats[(size_t)row0 * kFeatC + i];
  __syncthreads();

  v8f acc[4] = {};
  const int fr = lane & 15;
  for (int kk = 0; kk < kFeatC; kk += 4) {
    const int ko = kk + (hiL ? 2 : 0);
    v2f a;  // A 16x4 tile: lanes 0-15 K=kk,kk+1 ; lanes 16-31 K=kk+2,kk+3
    a.x = sF[wave][fr * kFeatC + ko];
    a.y = sF[wave][fr * kFeatC + ko + 1];
#pragma unroll
    for (int oc = 0; oc < 4; ++oc) {
      const int o = oc * 16 + fr;
      v2f b;  // B 4x16 tile: B[k][n] = Wout[o=n][c=k]
      b.x = sW[o * kFeatC + ko];
      b.y = sW[o * kFeatC + ko + 1];
      acc[oc] = __builtin_amdgcn_wmma_f32_16x16x4_f32(
          false, a, false, b, (short)0, acc[oc], false, false);
    }
  }

  // Epilogue: y = relu(acc * s + t), s = gamma*rsqrt(var+eps), t = beta-mean*s.
#pragma unroll
  for (int oc = 0; oc < 4; ++oc) {
    const int o = oc * 16 + fr;
    float s = bn_gamma[o] * rsqrtf(bn_var[o] + 1e-3f);
    float t = bn_beta[o] - bn_mean[o] * s;
#pragma unroll
    for (int r = 0; r < 8; ++r) {
      const int m = r + (hiL ? 8 : 0);     // C tile: VGPR r -> M = r (+8)
      float y = fmaxf(acc[oc][r] * s + t, 0.0f);
      out[(size_t)(row0 + m) * kOutC + o] = y;
    }
  }
}

// ---------------------------------------------------------------------------
extern "C" void kernel_launch(void* const* d_in, const int* in_sizes, int n_in,
                              void* d_out, int out_size, void* d_ws, size_t ws_size,
                              hipStream_t stream) {
  const float* lr_xyz   = (const float*)d_in[0];
  const float* lr_feat  = (const float*)d_in[1];
  const float* hr1_xyz  = (const float*)d_in[2];
  const float* hr1_feat = (const float*)d_in[3];
  const float* hr2_xyz  = (const float*)d_in[4];
  const float* hr2_feat = (const float*)d_in[5];
  const float* hr3_xyz  = (const float*)d_in[6];
  const float* hr3_feat = (const float*)d_in[7];
  const float* W14 = (const float*)d_in[8];
  const float* b14 = (const float*)d_in[9];
  const float* W24 = (const float*)d_in[10];
  const float* b24 = (const float*)d_in[11];
  const float* W34 = (const float*)d_in[12];
  const float* b34 = (const float*)d_in[13];
  const float* Wout = (const float*)d_in[14];
  const float* bn_gamma = (const float*)d_in[15];
  const float* bn_beta  = (const float*)d_in[16];
  const float* bn_mean  = (const float*)d_in[17];
  const float* bn_var   = (const float*)d_in[18];
  (void)in_sizes; (void)n_in; (void)out_size; (void)ws_size;

  float* feats = (float*)d_ws;  // [B*Nl, 128] fused channel matrix (4 MB)

  copy_lr_feat_kernel<<<(kBatch * kNl * 32 + 255) / 256, 256, 0, stream>>>(
      lr_feat, feats);

  dim3 grid(kNl / 16, kBatch);
  grouper_kernel<48, 2048><<<grid, 256, 0, stream>>>(
      lr_xyz, hr1_xyz, hr1_feat, W14, b14, feats, 16000, 32);
  grouper_kernel<32, 512><<<grid, 256, 0, stream>>>(
      lr_xyz, hr2_xyz, hr2_feat, W24, b24, feats, 8000, 80);
  grouper_kernel<16, 256><<<grid, 256, 0, stream>>>(
      lr_xyz, hr3_xyz, hr3_feat, W34, b34, feats, 4000, 112);

  out_gemm_kernel<<<(kBatch * kNl) / 64, 128, 0, stream>>>(
      feats, Wout, bn_gamma, bn_beta, bn_mean, bn_var, (float*)d_out);
}